// Conv_26104811225235
// MI455X (gfx1250) — compile-verified
//
#include <hip/hip_runtime.h>
#include <hip/hip_bf16.h>
#include <math.h>
#include <stdint.h>

// Segmented max over sorted segment ids (MinkowskiEngine-style pooling).
//   feats : [M, 60] float32, idx : [M] int32 (sorted), out : [S, 60] float32
// Bandwidth-bound: ~503 MB in / 0.98 MB out => ~22 us floor @ 23.3 TB/s.
// One workgroup per segment (contiguous range via binary search); stream the
// segment through LDS with CDNA5 async global->LDS copies, double-buffered
// via counted s_wait_asynccnt; reduce with v_max.
//
// The async VFLAT encoding adds the 24-bit instruction offset to BOTH the LDS
// and the global address, so full tiles need only one per-thread (la, ga) pair
// and 15 immediate-offset async issues -> near-zero address ALU in steady state.

#define TPTS 128   // points per tile
#define BLK  256   // 8 wave32 waves
#define DD   60    // feature dim

__device__ __forceinline__ int lower_bound_i32(const int* __restrict__ a, int n, int v) {
    int lo = 0, hi = n;
    while (lo < hi) {
        int mid = (lo + hi) >> 1;
        if (a[mid] < v) lo = mid + 1; else hi = mid;
    }
    return lo;
}

// One 8-byte async chunk; OFF (string literal) applies to LDS and global alike.
#define ASYNC_COPY8(la, ga, OFF)                                            \
    asm volatile("global_load_async_to_lds_b64 %0, %1, off offset:" OFF     \
                 :: "v"(la), "v"(ga) : "memory")

// Full tile: 3840 8B chunks / 256 threads = exactly 15 per wave.
__device__ __forceinline__ void issue_full_tile_v(uint32_t la, uint64_t ga) {
    ASYNC_COPY8(la, ga, "0");
    ASYNC_COPY8(la, ga, "2048");
    ASYNC_COPY8(la, ga, "4096");
    ASYNC_COPY8(la, ga, "6144");
    ASYNC_COPY8(la, ga, "8192");
    ASYNC_COPY8(la, ga, "10240");
    ASYNC_COPY8(la, ga, "12288");
    ASYNC_COPY8(la, ga, "14336");
    ASYNC_COPY8(la, ga, "16384");
    ASYNC_COPY8(la, ga, "18432");
    ASYNC_COPY8(la, ga, "20480");
    ASYNC_COPY8(la, ga, "22528");
    ASYNC_COPY8(la, ga, "24576");
    ASYNC_COPY8(la, ga, "26624");
    ASYNC_COPY8(la, ga, "28672");
}

// Partial (last) tile: clamp chunk id so every wave still issues exactly 15
// async instructions (keeps the counted-wait invariant); duplicate writes of
// identical data to the last valid slot are harmless.
__device__ __forceinline__ void issue_partial_tile(uint64_t gbase, int nch,
                                                   uint32_t ldsbase, int tid) {
#pragma unroll
    for (int j = 0; j < 15; ++j) {
        int c  = tid + (j << 8);
        int cc = (c < nch) ? c : (nch - 1);
        uint64_t ga = gbase   + ((uint64_t)(uint32_t)cc << 3);
        uint32_t la = ldsbase + ((uint32_t)cc << 3);
        ASYNC_COPY8(la, ga, "0");
    }
}

__global__ __launch_bounds__(BLK) void segmax60_kernel(
        const float* __restrict__ feats,
        const int*   __restrict__ idx,
        float*       __restrict__ out,
        int M) {

    __shared__ float buf0[TPTS * DD];   // 30,720 B
    __shared__ float buf1[TPTS * DD];   // 30,720 B   (double buffer)
    __shared__ float red[BLK];          //  1,024 B
    __shared__ int   rng[2];

    const int tid = threadIdx.x;
    const int seg = blockIdx.x;

    // Sorted ids => segment is a contiguous [start, end) range.
    if (tid < 2) rng[tid] = lower_bound_i32(idx, M, seg + tid);
    __syncthreads();
    const int start = rng[0];
    const int end   = rng[1];
    const int count = end - start;
    const int ntiles = (count + TPTS - 1) / TPTS;
    const int nfull  = count / TPTS;    // tiles with all TPTS points present

    // AS3 (LDS) byte offsets: low 32 bits of the flat pointer are the
    // wave-relative LDS address the async engine expects in VDST.
    const uint32_t tb   = (uint32_t)(tid << 3);
    const uint32_t lds0 = (uint32_t)(uintptr_t)(&buf0[0]);
    const uint32_t lds1 = (uint32_t)(uintptr_t)(&buf1[0]);
    const uint64_t gseg = (uint64_t)(uintptr_t)feats + (uint64_t)start * (DD * 4);

    const int x = tid & 63;   // feature dim slot (active when x < 60)
    const int y = tid >> 6;   // point stride lane (0..3)
    float m = -INFINITY;

    // Issue tile t into buffer (t&1): exactly 15 async instrs per wave.
    #define ISSUE(t)                                                          \
    do {                                                                      \
        const uint64_t _gt = gseg + (uint64_t)(t) * (TPTS * DD * 4);          \
        const uint32_t _lb = ((t) & 1) ? lds1 : lds0;                         \
        if ((t) < nfull) issue_full_tile_v(_lb + tb, _gt + tb);               \
        else issue_partial_tile(_gt, (count - (t) * TPTS) * (DD / 2),         \
                                _lb, tid);                                    \
    } while (0)

    if (ntiles > 0) ISSUE(0);

    for (int t = 0; t < ntiles; ++t) {
        if (t + 1 < ntiles) {
            // Prefetch next tile into the other buffer, then counted wait:
            // <=15 outstanding  =>  all of tile t's 15 ops (per wave) done.
            ISSUE(t + 1);
            asm volatile("s_wait_asynccnt 15" ::: "memory");
        } else {
            asm volatile("s_wait_asynccnt 0" ::: "memory");
        }
        __syncthreads();   // all 8 waves' portions of tile t are visible

        const float* lbuf = (t & 1) ? buf1 : buf0;
        int n = count - t * TPTS; if (n > TPTS) n = TPTS;
        if (x < DD) {
            for (int p = y; p < n; p += 4)
                m = fmaxf(m, lbuf[p * DD + x]);   // ds_load_b32 + v_max
        }
        __syncthreads();   // tile t buffer free before it is re-targeted
    }
    #undef ISSUE

    // Cross-y reduction (4 partial maxima per dim).
    red[tid] = m;
    __syncthreads();
    if (tid < 64) {
        float r = fmaxf(fmaxf(red[tid], red[tid + 64]),
                        fmaxf(red[tid + 128], red[tid + 192]));
        if (tid < DD)
            out[(size_t)seg * DD + tid] = (count > 0) ? r : 0.0f;
    }
}

// Generic fallback (any D) — correctness safety net, not the fast path.
__global__ void segmax_generic_kernel(const float* __restrict__ feats,
                                      const int*   __restrict__ idx,
                                      float*       __restrict__ out,
                                      int M, int D) {
    __shared__ int rng[2];
    const int seg = blockIdx.x;
    if (threadIdx.x < 2) rng[threadIdx.x] = lower_bound_i32(idx, M, seg + threadIdx.x);
    __syncthreads();
    const int start = rng[0], end = rng[1];
    for (int d = threadIdx.x; d < D; d += blockDim.x) {
        float m = -INFINITY;
        for (int p = start; p < end; ++p)
            m = fmaxf(m, feats[(size_t)p * D + d]);
        out[(size_t)seg * D + d] = (end > start) ? m : 0.0f;
    }
}

extern "C" void kernel_launch(void* const* d_in, const int* in_sizes, int n_in,
                              void* d_out, int out_size, void* d_ws, size_t ws_size,
                              hipStream_t stream) {
    (void)n_in; (void)d_ws; (void)ws_size;
    const float* feats = (const float*)d_in[0];
    const int*   idx   = (const int*)d_in[1];
    float*       out   = (float*)d_out;

    const int M = in_sizes[1];                      // total points
    const int D = (M > 0) ? (in_sizes[0] / M) : 0;  // feature dim (60)
    const int S = (D > 0) ? (out_size / D) : 0;     // segments (B*patch_num)
    if (S <= 0) return;

    if (D == DD) {
        segmax60_kernel<<<S, BLK, 0, stream>>>(feats, idx, out, M);
    } else {
        segmax_generic_kernel<<<S, 64, 0, stream>>>(feats, idx, out, M, D);
    }
}